// InitialTSDFModel_87883620810913
// MI455X (gfx1250) — compile-verified
//
#include <hip/hip_runtime.h>
#include <hip/hip_bf16.h>

// ---------------------------------------------------------------------------
// Eikonal signed-distance (64 Jacobi-Godunov sweeps) + minmax-normalize +
// bilinear x4 upsample, for B=4, H=W=512 float32.
//
// MI455X plan:
//  * 64 Jacobi sweeps fused 8-at-a-time with overlapped LDS tiling
//    (80x80 tile = 64x64 interior + halo 8, double-buffered in 51.2 KB of the
//    320 KB/WGP LDS). Exact Jacobi semantics for the interior; 8 launches
//    instead of 64 (launch latency was the dominant cost; compute is
//    L2-resident either way).
//  * Tile fill uses GLOBAL_LOAD_ASYNC_TO_LDS_B32 (ASYNCcnt) when the builtin
//    exists, else plain load + ds_store fallback.
//  * Bilinear x4 upsample as banded GEMMs on V_WMMA_F32_16X16X4_F32
//    (memory-bound stage; matrix pipe does the interpolation for free).
//    Output write (64 MB) is the roofline floor: ~3 us at 23.3 TB/s.
// ---------------------------------------------------------------------------

#define BIGV  1.0e6f
#define BATCH 4
#define HH    512
#define WW    512
#define IMG   (HH * WW)            // 262144
#define NPIX  (BATCH * IMG)        // 1048576
#define OH    (4 * HH)             // 2048
#define OW    (4 * WW)             // 2048
#define MIDIMG (OH * WW)           // 1048576 (row-upsampled, cols not yet)
#define NBLK  (NPIX / 256)         // 4096

// fused-sweep tiling
#define ITERS_PER_KERNEL 8
#define HALO  8
#define ITILE 64                   // interior tile edge (divides 512)
#define TS    (ITILE + 2 * HALO)   // 80
#define TCELL (TS * TS)            // 6400

typedef float v2f __attribute__((ext_vector_type(2)));
typedef float v8f __attribute__((ext_vector_type(8)));

// ---- optional CDNA5 async global->LDS path --------------------------------
#if defined(__gfx1250__) && __has_builtin(__builtin_amdgcn_global_load_async_to_lds_b32)
#define HAVE_ASYNC_LDS 1
typedef __attribute__((address_space(1))) int as1_int;
typedef __attribute__((address_space(3))) int as3_int;
__device__ __forceinline__ void async_g2l_b32(float* lds_dst, const float* gsrc) {
  // AS1 generic addresses are bit-identical; LDS generic addresses carry the
  // LDS byte offset in addr[31:0] (aperture rule), so int-casting is safe.
  __builtin_amdgcn_global_load_async_to_lds_b32(
      (as1_int*)(uintptr_t)gsrc,
      (as3_int*)(unsigned)(uintptr_t)lds_dst, 0, 0);
}
__device__ __forceinline__ void wait_async_lds() {
#if __has_builtin(__builtin_amdgcn_s_wait_asynccnt)
  __builtin_amdgcn_s_wait_asynccnt(0);
#else
  asm volatile("s_wait_asynccnt 0x0" ::: "memory");
#endif
}
#endif

// 8x4 bilinear weight band: 8 consecutive output rows (two x4 phases) vs the
// 4 input rows {2j-1, 2j, 2j+1, 2j+2}. Half-pixel centers, scale 4:
// phases r=0..3 -> taps (3/8,5/8),(1/8,7/8),(7/8,1/8),(5/8,3/8).
__device__ __constant__ float W84[8][4] = {
    {3.f/8.f, 5.f/8.f, 0.f,     0.f    },
    {1.f/8.f, 7.f/8.f, 0.f,     0.f    },
    {0.f,     7.f/8.f, 1.f/8.f, 0.f    },
    {0.f,     5.f/8.f, 3.f/8.f, 0.f    },
    {0.f,     3.f/8.f, 5.f/8.f, 0.f    },
    {0.f,     1.f/8.f, 7.f/8.f, 0.f    },
    {0.f,     0.f,     7.f/8.f, 1.f/8.f},
    {0.f,     0.f,     5.f/8.f, 3.f/8.f}};

// ---- 1) sub-pixel init of |distance| at the zero level set ---------------
__global__ __launch_bounds__(256) void k_init(const float* __restrict__ phi,
                                              float* __restrict__ d0) {
  int i = blockIdx.x * 256 + threadIdx.x;
  int x = i & (WW - 1);
  int y = (i / WW) & (HH - 1);
  float p = phi[i];
  float a = fabsf(p);
  float d = (p == 0.f) ? 0.f : BIGV;
  if (x < WW - 1) { float q = phi[i + 1];  if (p * q < 0.f) d = fminf(d, a / (a + fabsf(q))); }
  if (x > 0)      { float q = phi[i - 1];  if (p * q < 0.f) d = fminf(d, a / (a + fabsf(q))); }
  if (y < HH - 1) { float q = phi[i + WW]; if (p * q < 0.f) d = fminf(d, a / (a + fabsf(q))); }
  if (y > 0)      { float q = phi[i - WW]; if (p * q < 0.f) d = fminf(d, a / (a + fabsf(q))); }
  d0[i] = d;
}

// ---- 2) 8 fused Jacobi-Godunov sweeps via overlapped LDS tiling ----------
// Tile = 64x64 interior + halo 8 (80x80). After t in-LDS Jacobi iterations
// only cells >= t away from the tile edge are valid; with halo == iterations
// the 64x64 interior is bit-exact Jacobi. Out-of-image cells are re-clamped
// to BIG every iteration, matching the reference's per-sweep BIG padding.
__global__ __launch_bounds__(256) void k_godunov8(const float* __restrict__ din,
                                                  float* __restrict__ dout) {
  __shared__ float t0[TCELL];
  __shared__ float t1[TCELL];
  int tid = threadIdx.x;
  int b   = blockIdx.x >> 6;              // 64 tiles per image
  int rem = blockIdx.x & 63;
  int ty = rem >> 3, tx = rem & 7;
  int gx0 = tx * ITILE - HALO;
  int gy0 = ty * ITILE - HALO;
  const float* img = din + b * IMG;

  // ---- fill tile (async DMA to LDS where available) ----
  for (int idx = tid; idx < TCELL; idx += 256) {
    int ly = idx / TS, lx = idx - ly * TS;
    int gx = gx0 + lx, gy = gy0 + ly;
    bool in = (gx >= 0) && (gx < WW) && (gy >= 0) && (gy < HH);
#ifdef HAVE_ASYNC_LDS
    if (in) async_g2l_b32(&t0[idx], img + gy * WW + gx);
    else    t0[idx] = BIGV;
#else
    t0[idx] = in ? img[gy * WW + gx] : BIGV;
#endif
  }
#ifdef HAVE_ASYNC_LDS
  wait_async_lds();
#endif
  __syncthreads();

  // ---- 8 Jacobi iterations in LDS (double-buffered) ----
  float* cur = t0;
  float* nxt = t1;
#pragma unroll 1
  for (int t = 0; t < ITERS_PER_KERNEL; ++t) {
    for (int idx = tid; idx < TCELL; idx += 256) {
      int ly = idx / TS, lx = idx - ly * TS;
      float c  = cur[idx];
      float l  = (lx > 0)      ? cur[idx - 1]  : BIGV;
      float r  = (lx < TS - 1) ? cur[idx + 1]  : BIGV;
      float u  = (ly > 0)      ? cur[idx - TS] : BIGV;
      float dn = (ly < TS - 1) ? cur[idx + TS] : BIGV;
      float ax = fminf(l, r);
      float ay = fminf(u, dn);
      float mn = fminf(ax, ay);
      float df = fabsf(ax - ay);
      float two = 0.5f * (ax + ay + sqrtf(fmaxf(2.f - df * df, 0.f)));
      float cand = (df >= 1.f) ? (mn + 1.f) : two;
      int gx = gx0 + lx, gy = gy0 + ly;
      bool in = (gx >= 0) && (gx < WW) && (gy >= 0) && (gy < HH);
      nxt[idx] = in ? fminf(c, cand) : BIGV;
    }
    __syncthreads();
    float* tp = cur; cur = nxt; nxt = tp;
  }

  // ---- write the valid 64x64 interior ----
  float* outimg = dout + b * IMG;
  for (int idx = tid; idx < ITILE * ITILE; idx += 256) {
    int ly = idx >> 6, lx = idx & 63;
    int gx = gx0 + HALO + lx, gy = gy0 + HALO + ly;
    outimg[gy * WW + gx] = cur[(ly + HALO) * TS + (lx + HALO)];
  }
}

// ---- 3) apply sign, emit per-block min/max partials ----------------------
__global__ __launch_bounds__(256) void k_signmm(const float* __restrict__ phi,
                                                const float* __restrict__ d,
                                                float* __restrict__ sd,
                                                float* __restrict__ pmin,
                                                float* __restrict__ pmax) {
  __shared__ float smn[256];
  __shared__ float smx[256];
  int tid = threadIdx.x;
  int i = blockIdx.x * 256 + tid;
  float v = d[i];
  float s = (phi[i] < 0.f) ? -v : v;
  sd[i] = s;
  smn[tid] = s;
  smx[tid] = s;
  __syncthreads();
  for (int o = 128; o > 0; o >>= 1) {
    if (tid < o) {
      smn[tid] = fminf(smn[tid], smn[tid + o]);
      smx[tid] = fmaxf(smx[tid], smx[tid + o]);
    }
    __syncthreads();
  }
  if (tid == 0) { pmin[blockIdx.x] = smn[0]; pmax[blockIdx.x] = smx[0]; }
}

__global__ __launch_bounds__(256) void k_reduce(const float* __restrict__ pmin,
                                                const float* __restrict__ pmax,
                                                float* __restrict__ mm, int nb) {
  __shared__ float smn[256];
  __shared__ float smx[256];
  int tid = threadIdx.x;
  float mn = 2.f * BIGV, mx = -2.f * BIGV;
  for (int i = tid; i < nb; i += 256) {
    mn = fminf(mn, pmin[i]);
    mx = fmaxf(mx, pmax[i]);
  }
  smn[tid] = mn;
  smx[tid] = mx;
  __syncthreads();
  for (int o = 128; o > 0; o >>= 1) {
    if (tid < o) {
      smn[tid] = fminf(smn[tid], smn[tid + o]);
      smx[tid] = fmaxf(smx[tid], smx[tid + o]);
    }
    __syncthreads();
  }
  if (tid == 0) { mm[0] = smn[0]; mm[1] = smx[0]; }
}

// ---- 4) row upsample x4 via WMMA: D(16x16) = Wt(16x4) x In(4x16) ---------
__global__ __launch_bounds__(256) void k_rowup(const float* __restrict__ sd,
                                               float* __restrict__ mid) {
  int lane = threadIdx.x & 31;
  int wave = threadIdx.x >> 5;
  int t = blockIdx.x * 8 + wave;          // 32768 tiles total
  int b = t >> 13;
  int rem = t & 8191;
  int j = rem >> 5;                        // band 0..255
  int s = rem & 31;                        // 16-col strip
  int col0 = s * 16;
  int n = lane & 15;
  int kb = (lane >> 4) * 2;                // K pair: 0,1 (lanes<16) / 2,3

  v2f A;
  A.x = (n < 8) ? W84[n][kb]     : 0.f;
  A.y = (n < 8) ? W84[n][kb + 1] : 0.f;

  int base = 2 * j - 1;
  int r0 = min(max(base + kb,     0), HH - 1);
  int r1 = min(max(base + kb + 1, 0), HH - 1);
  const float* img = sd + b * IMG;
  v2f B;
  B.x = img[r0 * WW + col0 + n];
  B.y = img[r1 * WW + col0 + n];

  v8f C = {};
  C = __builtin_amdgcn_wmma_f32_16x16x4_f32(false, A, false, B, (short)0, C,
                                            false, false);

  if (lane < 16) {
    float* mo = mid + b * MIDIMG + (j * 8) * WW + col0 + lane;
#pragma unroll
    for (int v = 0; v < 8; ++v) mo[v * WW] = C[v];
  }
}

// ---- 5) col upsample x4 via WMMA + fused normalize -----------------------
__global__ __launch_bounds__(256) void k_colup(const float* __restrict__ mid,
                                               float* __restrict__ out,
                                               const float* __restrict__ mm) {
  int lane = threadIdx.x & 31;
  int wave = threadIdx.x >> 5;
  int t = blockIdx.x * 8 + wave;           // 131072 tiles total
  int b = t >> 15;
  int rem = t & 32767;
  int tr = rem >> 8;                        // 16-row chunk 0..127
  int j = rem & 255;                        // 8-col band
  int row0 = tr * 16;
  int oc0 = j * 8;
  int n = lane & 15;
  int kb = (lane >> 4) * 2;

  int base = 2 * j - 1;
  int c0 = min(max(base + kb,     0), WW - 1);
  int c1 = min(max(base + kb + 1, 0), WW - 1);
  const float* mi = mid + b * MIDIMG;
  v2f A;
  A.x = mi[(row0 + n) * WW + c0];
  A.y = mi[(row0 + n) * WW + c1];

  v2f B;
  B.x = (n < 8) ? W84[n][kb]     : 0.f;
  B.y = (n < 8) ? W84[n][kb + 1] : 0.f;

  v8f C = {};
  C = __builtin_amdgcn_wmma_f32_16x16x4_f32(false, A, false, B, (short)0, C,
                                            false, false);

  float mn = mm[0];
  float sc = 2.f / (mm[1] - mn);
  if (n < 8) {
    int rbase = (lane >> 4) * 8;
    float* oo = out + b * (OH * OW) + (row0 + rbase) * OW + oc0 + n;
#pragma unroll
    for (int v = 0; v < 8; ++v) oo[v * OW] = (C[v] - mn) * sc - 1.f;
  }
}

// ---------------------------------------------------------------------------
extern "C" void kernel_launch(void* const* d_in, const int* in_sizes, int n_in,
                              void* d_out, int out_size, void* d_ws,
                              size_t ws_size, hipStream_t stream) {
  const float* phi = (const float*)d_in[0];
  float* out = (float*)d_out;

  float* ws = (float*)d_ws;
  float* dA = ws;                               // NPIX
  float* dB = ws + NPIX;                        // NPIX
  float* mid = ws + 2 * NPIX;                   // BATCH*MIDIMG
  float* pmin = mid + BATCH * MIDIMG;           // NBLK
  float* pmax = pmin + NBLK;                    // NBLK
  float* mm = pmax + NBLK;                      // 2

  k_init<<<NBLK, 256, 0, stream>>>(phi, dA);

  float* a = dA;
  float* b = dB;
  for (int it = 0; it < 64 / ITERS_PER_KERNEL; ++it) {   // 8 launches, even count
    k_godunov8<<<BATCH * 64, 256, 0, stream>>>(a, b);
    float* tp = a; a = b; b = tp;
  }
  // result in dA (a), scratch in dB (b)

  k_signmm<<<NBLK, 256, 0, stream>>>(phi, a, b, pmin, pmax);  // sd -> b
  k_reduce<<<1, 256, 0, stream>>>(pmin, pmax, mm, NBLK);

  k_rowup<<<(BATCH * 256 * 32) / 8, 256, 0, stream>>>(b, mid);
  k_colup<<<(BATCH * 128 * 256) / 8, 256, 0, stream>>>(mid, out, mm);
}